// RelationGraphConv_70025146794668
// MI455X (gfx1250) — compile-verified
//
#include <hip/hip_runtime.h>
#include <hip/hip_bf16.h>
#include <math.h>

#define N_SRC 100000
#define N_DST 100000
#define NEDGE 1600000
#define D_IN  128
#define KH    8
#define DOUT  16
#define SLOPE 0.2f

typedef __attribute__((ext_vector_type(2))) float v2f;
typedef __attribute__((ext_vector_type(8))) float v8f;

__device__ __forceinline__ float leaky(float e) { return e >= 0.f ? e : SLOPE * e; }

// ordered-float atomic max via int/uint trick (init must be -inf)
__device__ __forceinline__ void atomicMaxF(float* addr, float v) {
    if (v >= 0.f) atomicMax((int*)addr, __float_as_int(v));
    else          atomicMin((unsigned int*)addr, __float_as_uint(v));
}

__global__ void __launch_bounds__(256)
fill_kernel(float* __restrict__ p, float v, int n) {
    int i = blockIdx.x * 256 + threadIdx.x;
    if (i < n) p[i] = v;
}

// attn[0..255] = feat_rel @ W_rel + b_rel   (one block of 256)
__global__ void attn_kernel(const float* __restrict__ feat_rel,
                            const float* __restrict__ W_rel,
                            const float* __restrict__ b_rel,
                            float* __restrict__ attn) {
    int j = threadIdx.x;
    float acc = b_rel[j];
    for (int i = 0; i < D_IN; ++i) acc += feat_rel[i] * W_rel[i * 2 * KH * DOUT + j];
    attn[j] = acc;
}

// W_eff[i][k] = sum_d W_dst[i][k*16+d] * attn[k*32+16+d];  cr[k] = bias term
__global__ void __launch_bounds__(256)
wreff_kernel(const float* __restrict__ W_dst, const float* __restrict__ b_dst,
             const float* __restrict__ attn,
             float* __restrict__ wr, float* __restrict__ cr) {
    int t = blockIdx.x * 256 + threadIdx.x;
    if (t < D_IN * KH) {
        int i = t >> 3, k = t & 7;
        float acc = 0.f;
        #pragma unroll
        for (int d = 0; d < DOUT; ++d)
            acc += W_dst[i * D_IN + k * DOUT + d] * attn[k * 2 * DOUT + DOUT + d];
        wr[i * KH + k] = acc;
    } else if (t < D_IN * KH + KH) {
        int k = t - D_IN * KH;
        float acc = 0.f;
        #pragma unroll
        for (int d = 0; d < DOUT; ++d)
            acc += b_dst[k * DOUT + d] * attn[k * 2 * DOUT + DOUT + d];
        cr[k] = acc;
    }
}

// fs = feat_src @ W_src + b_src  via V_WMMA_F32_16X16X4_F32
// one wave per 16x16 output tile; K=128 in steps of 4
__global__ void __launch_bounds__(128)
gemm_wmma_kernel(const float* __restrict__ A, const float* __restrict__ W,
                 const float* __restrict__ bias, float* __restrict__ out) {
    int wid  = threadIdx.x >> 5;
    int tile = blockIdx.x * 4 + wid;          // grid sized exactly: no OOB waves
    int row0 = (tile >> 3) * 16;
    int col0 = (tile & 7) * 16;
    int lane = threadIdx.x & 31;
    int half = lane >> 4, l16 = lane & 15;

    const float* Arow = A + (size_t)(row0 + l16) * D_IN + 2 * half;   // K pair base
    const float* Wcol = W + (size_t)(2 * half) * D_IN + col0 + l16;

    v8f c = {};
    #pragma unroll 4
    for (int k = 0; k < D_IN; k += 4) {
        v2f a, b;
        a.x = Arow[k];
        a.y = Arow[k + 1];
        b.x = Wcol[(size_t)k * D_IN];
        b.y = Wcol[(size_t)(k + 1) * D_IN];
        c = __builtin_amdgcn_wmma_f32_16x16x4_f32(
                /*neg_a=*/false, a, /*neg_b=*/false, b,
                /*c_mod=*/(short)0, c, /*reuse_a=*/false, /*reuse_b=*/false);
    }
    float bv = bias[col0 + l16];
    int rbase = row0 + half * 8;
    #pragma unroll
    for (int r = 0; r < 8; ++r)
        out[(size_t)(rbase + r) * D_IN + col0 + l16] = c[r] + bv;
}

// el[n][k] = sum_d fs[n][k*16+d] * attn[k*32+d]
__global__ void __launch_bounds__(256)
el_kernel(const float* __restrict__ fs, const float* __restrict__ attn,
          float* __restrict__ el) {
    int t = blockIdx.x * 256 + threadIdx.x;
    if (t >= N_SRC * KH) return;
    int node = t >> 3, k = t & 7;
    const float* f = fs + (size_t)node * D_IN + k * DOUT;
    const float* w = attn + k * 2 * DOUT;
    float acc = 0.f;
    #pragma unroll
    for (int d = 0; d < DOUT; ++d) acc += f[d] * w[d];
    el[t] = acc;
}

// er[n][k] = feat_dst[n] . W_eff[:,k] + cr[k]   (W_eff staged in LDS)
__global__ void __launch_bounds__(256)
er_kernel(const float* __restrict__ fd, const float* __restrict__ wr,
          const float* __restrict__ cr, float* __restrict__ er) {
    __shared__ float swr[D_IN * KH];
    __shared__ float scr[KH];
    for (int i = threadIdx.x; i < D_IN * KH; i += 256) swr[i] = wr[i];
    if (threadIdx.x < KH) scr[threadIdx.x] = cr[threadIdx.x];
    __syncthreads();
    int t = blockIdx.x * 256 + threadIdx.x;
    if (t >= N_DST * KH) return;
    int node = t >> 3, k = t & 7;
    const float* f = fd + (size_t)node * D_IN;
    float acc = scr[k];
    for (int i = 0; i < D_IN; ++i) acc += f[i] * swr[i * KH + k];
    er[t] = acc;
}

// pass 1: per-dst running max of leakyrelu scores
__global__ void __launch_bounds__(256)
edge_max_kernel(const int* __restrict__ src_idx, const int* __restrict__ dst_idx,
                const float* __restrict__ el, const float* __restrict__ er,
                float* __restrict__ dmax) {
    int eid = blockIdx.x * 256 + threadIdx.x;
    if (eid >= NEDGE) return;
    int s = src_idx[eid], d = dst_idx[eid];
    #pragma unroll
    for (int k = 0; k < KH; ++k) {
        float e = leaky(el[s * KH + k] + er[d * KH + k]);
        atomicMaxF(&dmax[d * KH + k], e);
    }
}

// pass 2: per-dst sum of exp(e - max)
__global__ void __launch_bounds__(256)
edge_sum_kernel(const int* __restrict__ src_idx, const int* __restrict__ dst_idx,
                const float* __restrict__ el, const float* __restrict__ er,
                const float* __restrict__ dmax, float* __restrict__ asum) {
    int eid = blockIdx.x * 256 + threadIdx.x;
    if (eid >= NEDGE) return;
    int s = src_idx[eid], d = dst_idx[eid];
    #pragma unroll
    for (int k = 0; k < KH; ++k) {
        float e = leaky(el[s * KH + k] + er[d * KH + k]);
        atomicAdd(&asum[d * KH + k], __expf(e - dmax[d * KH + k]));
    }
}

// pass 3: out[dst][j] += fs[src][j] * softmax_weight(head of j)
// thread per (edge, channel); 16 lanes share one head -> broadcast-friendly
__global__ void __launch_bounds__(256)
edge_scatter_kernel(const int* __restrict__ src_idx, const int* __restrict__ dst_idx,
                    const float* __restrict__ el, const float* __restrict__ er,
                    const float* __restrict__ dmax, const float* __restrict__ asum,
                    const float* __restrict__ fs, float* __restrict__ out) {
    int t = blockIdx.x * 256 + threadIdx.x;   // < NEDGE*128 = 204.8M, fits int
    int eid = t >> 7;
    int j = t & 127, k = j >> 4;
    int s = src_idx[eid], d = dst_idx[eid];
    float e = leaky(el[s * KH + k] + er[d * KH + k]);
    float a = __expf(e - dmax[d * KH + k]) / asum[d * KH + k];
    atomicAdd(&out[(size_t)d * D_IN + j], fs[(size_t)s * D_IN + j] * a);
}

extern "C" void kernel_launch(void* const* d_in, const int* in_sizes, int n_in,
                              void* d_out, int out_size, void* d_ws, size_t ws_size,
                              hipStream_t stream) {
    const float* feat_src = (const float*)d_in[0];
    const float* feat_dst = (const float*)d_in[1];
    const float* feat_rel = (const float*)d_in[2];
    const float* W_src    = (const float*)d_in[3];
    const float* b_src    = (const float*)d_in[4];
    const float* W_dst    = (const float*)d_in[5];
    const float* b_dst    = (const float*)d_in[6];
    const float* W_rel    = (const float*)d_in[7];
    const float* b_rel    = (const float*)d_in[8];
    const int*   src_idx  = (const int*)d_in[9];
    const int*   dst_idx  = (const int*)d_in[10];
    float* out = (float*)d_out;

    // workspace carve-up (~64 MB of floats)
    float* ws   = (float*)d_ws;
    float* fs   = ws; ws += (size_t)N_SRC * D_IN;   // 12.8M
    float* attn = ws; ws += 2 * KH * DOUT;          // 256
    float* wr   = ws; ws += D_IN * KH;              // 1024
    float* cr   = ws; ws += KH;                     // 8
    float* el   = ws; ws += (size_t)N_SRC * KH;     // 800K
    float* er   = ws; ws += (size_t)N_DST * KH;     // 800K
    float* dmax = ws; ws += (size_t)N_DST * KH;     // 800K
    float* asum = ws; ws += (size_t)N_DST * KH;     // 800K

    // init: out = 0, asum = 0, dmax = -inf
    fill_kernel<<<(N_DST * D_IN + 255) / 256, 256, 0, stream>>>(out, 0.f, N_DST * D_IN);
    fill_kernel<<<(N_DST * KH + 255) / 256, 256, 0, stream>>>(asum, 0.f, N_DST * KH);
    fill_kernel<<<(N_DST * KH + 255) / 256, 256, 0, stream>>>(dmax, -INFINITY, N_DST * KH);

    // tiny projections
    attn_kernel<<<1, 256, 0, stream>>>(feat_rel, W_rel, b_rel, attn);
    wreff_kernel<<<5, 256, 0, stream>>>(W_dst, b_dst, attn, wr, cr);

    // fs = feat_src @ W_src + b_src   (WMMA fp32; 6250*8 tiles, 4 waves/block)
    gemm_wmma_kernel<<<(N_SRC / 16) * 8 / 4, 128, 0, stream>>>(feat_src, W_src, b_src, fs);

    // attention logits per node
    el_kernel<<<(N_SRC * KH + 255) / 256, 256, 0, stream>>>(fs, attn, el);
    er_kernel<<<(N_DST * KH + 255) / 256, 256, 0, stream>>>(feat_dst, wr, cr, er);

    // edge softmax (3 passes, scores recomputed from L2-resident el/er)
    edge_max_kernel<<<(NEDGE + 255) / 256, 256, 0, stream>>>(src_idx, dst_idx, el, er, dmax);
    edge_sum_kernel<<<(NEDGE + 255) / 256, 256, 0, stream>>>(src_idx, dst_idx, el, er, dmax, asum);
    edge_scatter_kernel<<<NEDGE / 2, 256, 0, stream>>>(src_idx, dst_idx, el, er, dmax, asum, fs, out);
}